// AwareDecoder_84232898609641
// MI455X (gfx1250) — compile-verified
//
#include <hip/hip_runtime.h>
#include <hip/hip_bf16.h>

// Problem dims from the reference (compile-time for full unrolling / static LDS).
#define BB 32
#define SS 2048
#define HH 1024
#define JJ 20
#define BLOCK 256

// Pointee type the gfx1250 async builtins expect: 16-byte int vector.
typedef int v4i __attribute__((vector_size(16)));
typedef __attribute__((address_space(1))) v4i gbl_v4i;  // global
typedef __attribute__((address_space(3))) v4i lds_v4i;  // LDS

// ---------------------------------------------------------------------------
// CDNA5 async LDS DMA helpers (ASYNCcnt-tracked, per-lane 16B transfers).
// Prefer clang builtins when present; fall back to inline gfx1250 asm.
// ---------------------------------------------------------------------------
__device__ __forceinline__ void async_load16(const float* g, float* lds) {
#if __has_builtin(__builtin_amdgcn_global_load_async_to_lds_b128)
    __builtin_amdgcn_global_load_async_to_lds_b128(
        (gbl_v4i*)(float*)g,
        (lds_v4i*)lds,
        0, 0);
#else
    unsigned lds_addr =
        (unsigned)(__SIZE_TYPE__)(__attribute__((address_space(3))) void*)lds;
    asm volatile("global_load_async_to_lds_b128 %0, %1, off"
                 :
                 : "v"(lds_addr), "v"(g)
                 : "memory");
#endif
}

__device__ __forceinline__ void async_store16(float* g, const float* lds) {
#if __has_builtin(__builtin_amdgcn_global_store_async_from_lds_b128)
    __builtin_amdgcn_global_store_async_from_lds_b128(
        (gbl_v4i*)g,
        (lds_v4i*)(float*)lds,
        0, 0);
#else
    unsigned lds_addr =
        (unsigned)(__SIZE_TYPE__)(__attribute__((address_space(3))) void*)(float*)lds;
    asm volatile("global_store_async_from_lds_b128 %0, %1, off"
                 :
                 : "v"(g), "v"(lds_addr)
                 : "memory");
#endif
}

__device__ __forceinline__ void wait_async0() {
#if __has_builtin(__builtin_amdgcn_s_wait_asynccnt)
    __builtin_amdgcn_s_wait_asynccnt(0);
#else
    asm volatile("s_wait_asynccnt 0" ::: "memory");
#endif
}

// ---------------------------------------------------------------------------
// One block per (batch, label). 640 blocks x 256 threads.
//   Phase 1: scan mask row (int4 vectorized), min/max reduce in LDS.
//   Phase 2: DMA-gather first/last hidden rows global->LDS (async),
//            then DMA LDS->output row (async). Zero-fill if label absent.
// ---------------------------------------------------------------------------
__global__ void __launch_bounds__(BLOCK)
aware_decoder_gather(const float* __restrict__ input,      // [B,S,H]
                     const int*   __restrict__ mask,       // [B,S]
                     const int*   __restrict__ max_number, // scalar
                     float*       __restrict__ out) {      // [B,J,2H]
    __shared__ int sFirst;
    __shared__ int sLast;
    __shared__ __align__(16) float tile[2 * HH];  // 8 KB staging

    const int bj    = blockIdx.x;
    const int b     = bj / JJ;
    const int j     = bj % JJ;
    const int label = j + 1;
    const int tid   = threadIdx.x;

    if (tid == 0) { sFirst = SS; sLast = -1; }
    __syncthreads();

    // ---- Phase 1: scan mask row for `label` ----
    int localF = SS;
    int localL = -1;
    if (label <= max_number[0]) {
        const int4* row4 = reinterpret_cast<const int4*>(mask + (size_t)b * SS);
        #pragma unroll
        for (int i = tid; i < SS / 4; i += BLOCK) {
            const int4 v = row4[i];
            const int p = i << 2;
            if (v.x == label) { localF = min(localF, p + 0); localL = max(localL, p + 0); }
            if (v.y == label) { localF = min(localF, p + 1); localL = max(localL, p + 1); }
            if (v.z == label) { localF = min(localF, p + 2); localL = max(localL, p + 2); }
            if (v.w == label) { localF = min(localF, p + 3); localL = max(localL, p + 3); }
        }
    }
    if (localF < SS) atomicMin(&sFirst, localF);
    if (localL >= 0) atomicMax(&sLast, localL);
    __syncthreads();

    const int f = sFirst;
    const int l = sLast;
    float* outRow = out + (size_t)bj * (2 * HH);

    // ---- Absent label: vectorized zero-fill of the output row ----
    if (l < 0) {
        float4  z  = make_float4(0.f, 0.f, 0.f, 0.f);
        float4* o4 = reinterpret_cast<float4*>(outRow);
        #pragma unroll
        for (int i = tid; i < (2 * HH) / 4; i += BLOCK) o4[i] = z;
        return;
    }

    // ---- Phase 2: DMA gather through LDS ----
    const float* srcF = input + ((size_t)b * SS + (size_t)f) * HH;
    const float* srcL = input + ((size_t)b * SS + (size_t)l) * HH;

    // HH/4 == 256 == BLOCK: exactly one b128 per thread per row.
    #pragma unroll
    for (int i = tid; i < HH / 4; i += BLOCK) {
        async_load16(srcF + 4 * i, tile + 4 * i);
        async_load16(srcL + 4 * i, tile + HH + 4 * i);
    }
    wait_async0();       // this wave's loads landed in LDS
    __syncthreads();     // all waves' loads landed

    // (2*HH)/4 == 512: two b128 stores per thread, straight from LDS.
    #pragma unroll
    for (int i = tid; i < (2 * HH) / 4; i += BLOCK) {
        async_store16(outRow + 4 * i, tile + 4 * i);
    }
    wait_async0();       // drain before wave retires (ENDPGM waits anyway)
}

extern "C" void kernel_launch(void* const* d_in, const int* in_sizes, int n_in,
                              void* d_out, int out_size, void* d_ws, size_t ws_size,
                              hipStream_t stream) {
    (void)in_sizes; (void)n_in; (void)d_ws; (void)ws_size; (void)out_size;
    const float* input      = (const float*)d_in[0];
    const int*   mask       = (const int*)d_in[1];
    const int*   max_number = (const int*)d_in[2];
    float*       out        = (float*)d_out;

    dim3 grid(BB * JJ);   // 640 blocks, one per (batch, label)
    dim3 block(BLOCK);    // 8 wave32s
    aware_decoder_gather<<<grid, block, 0, stream>>>(input, mask, max_number, out);
}